// ProposalInductionLoss_34256659153158
// MI455X (gfx1250) — compile-verified
//
#include <hip/hip_runtime.h>

// ---------------------------------------------------------------------------
// ProposalInductionLoss, fused for MI455X (gfx1250, wave32).
//
//   loss = mean( max(x,0) - x*z + log1p(exp(-|x|)) )
// with z a per-(b,c) one-hot at the mask centroid. Since z is one-hot,
//   N*loss = sum_x softplus_term(x) - sum_slices x[centroid]
//
// Pass 1: stream pred+mask once (256 MiB, HBM-bound ~11us @ 23.3 TB/s) with
//         b128 NT loads + global_prefetch_b8. Per-thread accumulators are
//         reduced across each wave on the MATRIX pipe: V_WMMA_F32_16X16X4_F32
//         with B=ones sums the 32 lane values (all four quantities go through
//         WMMA; the integer stats are exact in fp32: wave totals < 2^24).
// Pass 2: 1 block: centroids, 128 scattered gathers, fixed-order reduce.
// Determinism: integer atomics + fixed-slot partials + fixed-order trees.
// ---------------------------------------------------------------------------

typedef __attribute__((ext_vector_type(2))) float v2f;
typedef __attribute__((ext_vector_type(4))) float v4f;
typedef __attribute__((ext_vector_type(8))) float v8f;

#define NSLICE 128            // B*C = 16*8
#define BLOCKS_PER_SLICE 16
#define NBLOCKS (NSLICE * BLOCKS_PER_SLICE)   // 2048
#define THREADS 256
#define VEC_PER_THREAD 16     // float4 per thread -> 64 elems/thread

#define LOG2E 1.4426950408889634f
#define LN2   0.6931471805599453f

// Reduce one fp32 value across the wave using the WMMA pipe:
// A(16x4) holds the 32 lane values (a.y = 0 pads K), B(4x16) = ones, so
// D[m][n] = v_m + v_{m+16}. Lanes 0-15 sum rows 0-7 of their column, lanes
// 16-31 sum rows 8-15; one shfl_xor(16) completes the wave sum (all lanes).
__device__ __forceinline__ float wave_sum_wmma(float v) {
  v2f a; a.x = v;    a.y = 0.0f;
  v2f b; b.x = 1.0f; b.y = 1.0f;
  v8f c = {};
  v8f d = __builtin_amdgcn_wmma_f32_16x16x4_f32(
      /*neg_a=*/false, a, /*neg_b=*/false, b,
      /*c_mod=*/(short)0, c, /*reuse_a=*/false, /*reuse_b=*/false);
  float s = d[0] + d[1] + d[2] + d[3] + d[4] + d[5] + d[6] + d[7];
  s += __shfl_xor(s, 16, 32);
  return s;
}

__global__ __launch_bounds__(THREADS)
void pil_pass1(const float* __restrict__ pred, const float* __restrict__ mask,
               unsigned* __restrict__ g_cnt, unsigned* __restrict__ g_rsum,
               unsigned* __restrict__ g_csum, float* __restrict__ partials) {
  const int slice = blockIdx.x >> 4;          // 16 blocks per slice
  const int chunk = blockIdx.x & 15;
  const int tid   = threadIdx.x;

  const v4f* __restrict__ pred4 = (const v4f*)pred;
  const v4f* __restrict__ mask4 = (const v4f*)mask;
  const long sliceVecBase = ((long)slice) << 16;   // 65536 float4 per slice

  float acc_lin = 0.0f;   // sum of max(x,0)
  float acc_log = 0.0f;   // sum of log2(1 + 2^(-|x|*log2e)); *ln2 folded out
  unsigned tc = 0, tr = 0, ts = 0;

#pragma unroll
  for (int i = 0; i < VEC_PER_THREAD; ++i) {
    const int  v  = (chunk << 12) + (i << 8) + tid; // vec idx within slice
    const long gv = sliceVecBase + v;
    if (i < VEC_PER_THREAD - 8) {                   // stay inside this chunk
      __builtin_prefetch(&pred4[gv + (8 << 8)], 0, 0); // -> global_prefetch_b8
      __builtin_prefetch(&mask4[gv + (8 << 8)], 0, 0);
    }
    v4f p = __builtin_nontemporal_load(&pred4[gv]); // streaming, NT hint
    v4f m = __builtin_nontemporal_load(&mask4[gv]);

    // Raw HW transcendentals: exp2 input <= 0, log2 input in [1,2] -> no
    // libm range fixups needed (bare v_exp_f32 / v_log_f32).
#pragma unroll
    for (int j = 0; j < 4; ++j) {
      const float x  = p[j];
      const float ax = __builtin_fabsf(x);
      acc_lin += fmaxf(x, 0.0f);
      acc_log += __builtin_amdgcn_logf(1.0f + __builtin_amdgcn_exp2f(ax * -LOG2E));
    }

    const int e = v << 2;                 // element idx within 512x512 slice
    const unsigned r  = (unsigned)(e >> 9);
    const unsigned c0 = (unsigned)(e & 511);
    const unsigned w0 = (m.x != 0.0f), w1 = (m.y != 0.0f);
    const unsigned w2 = (m.z != 0.0f), w3 = (m.w != 0.0f);
    const unsigned k = w0 + w1 + w2 + w3;
    tc += k;
    tr += k * r;
    ts += w0 * c0 + w1 * (c0 + 1u) + w2 * (c0 + 2u) + w3 * (c0 + 3u);
  }

  const float acc = fmaf(LN2, acc_log, acc_lin);

  // All four wave reductions on the matrix pipe (integer stats are exact in
  // fp32: per-wave totals < 2^24). Avoids 15 rounds of ds_bpermute.
  const float wsf = wave_sum_wmma(acc);
  const float wcf = wave_sum_wmma((float)tc);
  const float wrf = wave_sum_wmma((float)tr);
  const float wzf = wave_sum_wmma((float)ts);

  __shared__ float    sF[THREADS / 32];
  __shared__ unsigned sC[THREADS / 32], sR[THREADS / 32], sS[THREADS / 32];
  const int wave = tid >> 5, lane = tid & 31;
  if (lane == 0) {
    sF[wave] = wsf;
    sC[wave] = (unsigned)wcf;   // exact integer-valued floats
    sR[wave] = (unsigned)wrf;
    sS[wave] = (unsigned)wzf;
  }
  __syncthreads();
  if (tid == 0) {
    float f = 0.0f; unsigned a = 0, b = 0, c = 0;
#pragma unroll
    for (int w = 0; w < THREADS / 32; ++w) { f += sF[w]; a += sC[w]; b += sR[w]; c += sS[w]; }
    partials[blockIdx.x] = f;                 // fixed slot -> deterministic
    atomicAdd(&g_cnt[slice],  a);             // integer atomics: deterministic
    atomicAdd(&g_rsum[slice], b);
    atomicAdd(&g_csum[slice], c);
  }
}

__global__ __launch_bounds__(256)
void pil_pass2(const float* __restrict__ pred,
               const unsigned* __restrict__ g_cnt,
               const unsigned* __restrict__ g_rsum,
               const unsigned* __restrict__ g_csum,
               const float* __restrict__ partials,
               float* __restrict__ out) {
  const int t = threadIdx.x;

  // Fixed-order sum of the 2048 per-block partials (8 per thread).
  float ps = 0.0f;
#pragma unroll
  for (int k = 0; k < NBLOCKS / 256; ++k) ps += partials[t + (k << 8)];

  // Centroid gather term for slices 0..127.
  float g = 0.0f;
  if (t < NSLICE) {
    const unsigned n = g_cnt[t];
    if (n > 0u) {
      const float fn = (float)n;
      const int mr = (int)((float)g_rsum[t] / fn);  // trunc, coords >= 0
      const int mc = (int)((float)g_csum[t] / fn);
      g = pred[(((long)t) << 18) + ((long)mr << 9) + mc];
    }
  }

  __shared__ float red[256];
  red[t] = ps - g;
  __syncthreads();
#pragma unroll
  for (int s = 128; s > 0; s >>= 1) {      // fixed tree -> deterministic
    if (t < s) red[t] += red[t + s];
    __syncthreads();
  }
  if (t == 0) out[0] = red[0] * 2.9802322387695312e-08f;  // * 2^-25 (exact)
}

extern "C" void kernel_launch(void* const* d_in, const int* in_sizes, int n_in,
                              void* d_out, int out_size, void* d_ws, size_t ws_size,
                              hipStream_t stream) {
  const float* pred = (const float*)d_in[0];
  const float* mask = (const float*)d_in[1];

  unsigned* cnt  = (unsigned*)d_ws;
  unsigned* rsum = cnt + NSLICE;
  unsigned* csum = cnt + 2 * NSLICE;
  float* partials = (float*)(cnt + 3 * NSLICE);   // NBLOCKS floats, ~10KB total

  hipMemsetAsync(d_ws, 0, 3 * NSLICE * sizeof(unsigned), stream);
  pil_pass1<<<NBLOCKS, THREADS, 0, stream>>>(pred, mask, cnt, rsum, csum, partials);
  pil_pass2<<<1, 256, 0, stream>>>(pred, cnt, rsum, csum, partials, (float*)d_out);
}